// KGNN_8246337208547
// MI455X (gfx1250) — compile-verified
//
#include <hip/hip_runtime.h>

#define N_NODES 50000
#define N_EDGES 800000
#define IN_C 128
#define HID_C 128
#define OUT_C 64
#define BN_EPS 1e-5f

// 50000 rows = 3125 16-row tiles = 625 groups of 5 tiles (3125 = 5^5)
#define MT 5          // M-subtiles per wave
#define MROWS (MT*16) // 80 rows per wave
#define MGROUPS (N_NODES / MROWS) // 625

typedef __attribute__((ext_vector_type(2))) float v2f;
typedef __attribute__((ext_vector_type(8))) float v8f;

// ---------------------------------------------------------------------------
// Scatter: agg[dst] += w * x[src].  One wave per edge; each lane handles 4
// channels with a float4 gather + 4 global_atomic_add_f32 (resolved in L2,
// since the 25.6MB feature table fits the 192MB L2).
// ---------------------------------------------------------------------------
__global__ void scatter_kernel(const float* __restrict__ x,
                               const int* __restrict__ src,
                               const int* __restrict__ dst,
                               const float* __restrict__ ew,
                               float* __restrict__ agg) {
    int wid  = (blockIdx.x * blockDim.x + threadIdx.x) >> 5;   // edge id
    int lane = threadIdx.x & 31;
    if (wid >= N_EDGES) return;
    int   s = src[wid];
    int   d = dst[wid];
    float w = ew[wid];
    const float4* xs = (const float4*)(x + (size_t)s * IN_C);
    float4 v = xs[lane];                                       // 4 channels
    float* ad = agg + (size_t)d * IN_C + lane * 4;
    atomicAdd(ad + 0, w * v.x);
    atomicAdd(ad + 1, w * v.y);
    atomicAdd(ad + 2, w * v.z);
    atomicAdd(ad + 3, w * v.w);
}

// ---------------------------------------------------------------------------
// Fused dual GEMM:  out = A1 @ W1 + A2 @ W2 + bias   (A: N x 128, W: 128 x NC)
// One wave owns a 80x16 output strip (5 WMMA M-subtiles x one N-tile).
// Per K-step: 1 B-fragment load, reused by 5 WMMAs (5x less weight traffic).
// V_WMMA_F32_16X16X4_F32, K=128 -> 2*5*32 = 320 WMMAs per wave.
// Optionally accumulates per-channel sum / sum-of-squares for BatchNorm.
//
// Fragment layouts per CDNA5 ISA 7.12.2:
//   A (16x4):  lane = (k>=2)*16 + m ; VGPR0=K even, VGPR1=K odd  -> v2f/lane
//   B (4x16):  lane = (k>=2)*16 + n ; mirrored                    -> v2f/lane
//   C (16x16): VGPR v, lanes 0-15: M=v,N=lane; lanes 16-31: M=v+8,N=lane-16
// ---------------------------------------------------------------------------
template <int NC, bool STATS>
__global__ void gemm_wmma_kernel(const float* __restrict__ A1,
                                 const float* __restrict__ A2,
                                 const float* __restrict__ W1,
                                 const float* __restrict__ W2,
                                 const float* __restrict__ bias,
                                 float* __restrict__ out,
                                 float* __restrict__ sum,
                                 float* __restrict__ sumsq) {
    const int ntiles = NC / 16;
    const int groups = MGROUPS * ntiles;
    int wid  = blockIdx.x * (blockDim.x >> 5) + (threadIdx.x >> 5);
    int lane = threadIdx.x & 31;
    if (wid >= groups) return;                // uniform per wave -> EXEC all-1s

    int mg  = wid / ntiles;
    int nt  = wid % ntiles;
    int row0 = mg * MROWS + (lane & 15);      // first A row (M) of subtile 0
    int col  = nt * 16 + (lane & 15);         // B/C col (N)
    int kb   = (lane >> 4) * 2;               // 0 or 2 (K sub-offset per lane)

    const float* w1p = W1 + (size_t)kb * NC + col;
    const float* w2p = W2 + (size_t)kb * NC + col;

    v8f acc[MT];
#pragma unroll
    for (int t = 0; t < MT; ++t) acc[t] = (v8f){};

    // ---- GEMM 1: agg @ W_rel ----
    {
        const float* ap = A1 + (size_t)row0 * 128 + kb;
#pragma unroll 4
        for (int k = 0; k < 128; k += 4) {
            v2f b;
            b[0] = w1p[(size_t)k * NC];
            b[1] = w1p[(size_t)(k + 1) * NC];
#pragma unroll
            for (int t = 0; t < MT; ++t) {
                v2f a;
                a[0] = ap[(size_t)t * 16 * 128 + k];
                a[1] = ap[(size_t)t * 16 * 128 + k + 1];
                acc[t] = __builtin_amdgcn_wmma_f32_16x16x4_f32(
                    false, a, false, b, (short)0, acc[t], false, false);
            }
        }
    }
    // ---- GEMM 2: x @ W_root ----
    {
        const float* ap = A2 + (size_t)row0 * 128 + kb;
#pragma unroll 4
        for (int k = 0; k < 128; k += 4) {
            v2f b;
            b[0] = w2p[(size_t)k * NC];
            b[1] = w2p[(size_t)(k + 1) * NC];
#pragma unroll
            for (int t = 0; t < MT; ++t) {
                v2f a;
                a[0] = ap[(size_t)t * 16 * 128 + k];
                a[1] = ap[(size_t)t * 16 * 128 + k + 1];
                acc[t] = __builtin_amdgcn_wmma_f32_16x16x4_f32(
                    false, a, false, b, (short)0, acc[t], false, false);
            }
        }
    }

    float bc = bias[col];
    float s = 0.f, ss = 0.f;
    int mhalf = (lane >> 4) * 8;              // C rows: v + 8*(lane>=16)
#pragma unroll
    for (int t = 0; t < MT; ++t) {
        int rbase = mg * MROWS + t * 16 + mhalf;
#pragma unroll
        for (int v = 0; v < 8; ++v) {
            float val = acc[t][v] + bc;
            out[(size_t)(rbase + v) * NC + col] = val;
            if (STATS) { s += val; ss += val * val; }
        }
    }
    if (STATS) {
        // combine the two half-wave partials for the same column, then one
        // atomic per column from the low half (32 stat-atomics per wave).
        s  += __shfl_xor(s, 16, 32);
        ss += __shfl_xor(ss, 16, 32);
        if (lane < 16) {
            atomicAdd(&sum[col], s);
            atomicAdd(&sumsq[col], ss);
        }
    }
}

// ---------------------------------------------------------------------------
// BN stat finalize: musig[c] = mean, musig[128+c] = rsqrt(var + eps)
// ---------------------------------------------------------------------------
__global__ void bnstats_kernel(const float* __restrict__ sum,
                               const float* __restrict__ sumsq,
                               float* __restrict__ musig) {
    int c = threadIdx.x;                       // 128 threads
    float inv_n = 1.0f / (float)N_NODES;
    float mu  = sum[c] * inv_n;
    float var = sumsq[c] * inv_n - mu * mu;
    musig[c]         = mu;
    musig[HID_C + c] = rsqrtf(var + BN_EPS);
}

// ---------------------------------------------------------------------------
// BN apply + residual + ReLU:  xout = relu((h-mu)*rs*gamma + beta + xin)
// Pure elementwise at identical index -> safe for xin == xout (in place).
// ---------------------------------------------------------------------------
__global__ void bnapply_kernel(const float* __restrict__ h,
                               const float* __restrict__ xin,
                               const float* __restrict__ musig,
                               const float* __restrict__ gamma,
                               const float* __restrict__ beta,
                               float* __restrict__ xout) {
    size_t i = (size_t)blockIdx.x * blockDim.x + threadIdx.x;
    if (i >= (size_t)N_NODES * HID_C) return;
    int ch = (int)(i & (HID_C - 1));
    float mu = musig[ch];
    float rs = musig[HID_C + ch];
    float v  = (h[i] - mu) * rs * gamma[ch] + beta[ch] + xin[i];
    xout[i]  = fmaxf(v, 0.0f);
}

// ---------------------------------------------------------------------------
extern "C" void kernel_launch(void* const* d_in, const int* in_sizes, int n_in,
                              void* d_out, int out_size, void* d_ws, size_t ws_size,
                              hipStream_t stream) {
    const float* x    = (const float*)d_in[0];
    const int*   esrc = (const int*)d_in[1];
    const int*   edst = (const int*)d_in[2];
    const float* ew   = (const float*)d_in[3];
    const float* Wr0  = (const float*)d_in[4];
    const float* br0  = (const float*)d_in[5];
    const float* Wo0  = (const float*)d_in[6];
    const float* Wr1  = (const float*)d_in[7];
    const float* br1  = (const float*)d_in[8];
    const float* Wo1  = (const float*)d_in[9];
    const float* Wr2  = (const float*)d_in[10];
    const float* br2  = (const float*)d_in[11];
    const float* Wo2  = (const float*)d_in[12];
    const float* g0   = (const float*)d_in[13];
    const float* be0  = (const float*)d_in[14];
    const float* g1   = (const float*)d_in[15];
    const float* be1  = (const float*)d_in[16];
    float* out = (float*)d_out;

    // workspace layout: agg | h | xb | sum | sumsq | musig
    const size_t NF = (size_t)N_NODES * HID_C;
    float* agg   = (float*)d_ws;
    float* h     = agg + NF;
    float* xb    = h + NF;
    float* sum   = xb + NF;
    float* ssq   = sum + HID_C;
    float* musig = ssq + HID_C;

    const int scatterBlocks = (N_EDGES * 32) / 256;                  // 100000
    const int gemmBlocks128 = (MGROUPS * (HID_C / 16) + 7) / 8;      // 625
    const int gemmBlocks64  = (MGROUPS * (OUT_C / 16) + 7) / 8;      // 313
    const int ewBlocks      = (int)((NF + 255) / 256);               // 25000

    // ---------------- layer 0 ----------------
    hipMemsetAsync(agg, 0, NF * sizeof(float), stream);
    hipMemsetAsync(sum, 0, 2 * HID_C * sizeof(float), stream);
    scatter_kernel<<<scatterBlocks, 256, 0, stream>>>(x, esrc, edst, ew, agg);
    gemm_wmma_kernel<HID_C, true><<<gemmBlocks128, 256, 0, stream>>>(
        agg, x, Wr0, Wo0, br0, h, sum, ssq);
    bnstats_kernel<<<1, HID_C, 0, stream>>>(sum, ssq, musig);
    bnapply_kernel<<<ewBlocks, 256, 0, stream>>>(h, x, musig, g0, be0, xb);

    // ---------------- layer 1 ----------------
    hipMemsetAsync(agg, 0, NF * sizeof(float), stream);
    hipMemsetAsync(sum, 0, 2 * HID_C * sizeof(float), stream);
    scatter_kernel<<<scatterBlocks, 256, 0, stream>>>(xb, esrc, edst, ew, agg);
    gemm_wmma_kernel<HID_C, true><<<gemmBlocks128, 256, 0, stream>>>(
        agg, xb, Wr1, Wo1, br1, h, sum, ssq);
    bnstats_kernel<<<1, HID_C, 0, stream>>>(sum, ssq, musig);
    bnapply_kernel<<<ewBlocks, 256, 0, stream>>>(h, xb, musig, g1, be1, xb);

    // ---------------- layer 2 (no BN / residual) ----------------
    hipMemsetAsync(agg, 0, NF * sizeof(float), stream);
    scatter_kernel<<<scatterBlocks, 256, 0, stream>>>(xb, esrc, edst, ew, agg);
    gemm_wmma_kernel<OUT_C, false><<<gemmBlocks64, 256, 0, stream>>>(
        agg, xb, Wr2, Wo2, br2, out, nullptr, nullptr);
}